// Discriminator_71098888618114
// MI455X (gfx1250) — compile-verified
//
#include <hip/hip_runtime.h>

typedef __attribute__((ext_vector_type(2))) float v2f;
typedef __attribute__((ext_vector_type(8))) float v8f;

namespace {
constexpr int D              = 1024;
constexpr int ROWS_PER_BLOCK = 128;   // 8 waves * 16 rows
constexpr int EB             = 128;   // e-columns per outer block = 8 WMMA tiles
constexpr int KS             = 64;    // k-slab staged in LDS
constexpr int KPAD           = 68;    // LDS row stride (floats): bank = (4e + k) % 64
                                      // -> B-frag reads conflict-free (halves use disjoint banks)
}

__global__ __launch_bounds__(256, 2)
void bilinear_wmma_f32_kernel(const float* __restrict__ elg,
                              const float* __restrict__ eth,
                              const float* __restrict__ W,
                              float* __restrict__ out)
{
    __shared__ float sW[EB * KPAD];   // 34,816 bytes, layout [e][k] (transposed)

    const int tid  = threadIdx.x;
    const int wave = tid >> 5;
    const int lane = tid & 31;
    const int colg = lane & 15;            // N column within 16-tile; also M row for A frag
    const int kh   = (lane >> 4) << 1;     // 0 or 2 : K sub-offset for A/B fragments

    const int rowTile = blockIdx.x * ROWS_PER_BLOCK + wave * 16;
    const float* aRow = elg + (size_t)(rowTile + colg) * D;

    float acc[8];
#pragma unroll
    for (int r = 0; r < 8; ++r) acc[r] = 0.f;

    for (int eb = 0; eb < D / EB; ++eb) {
        const int e0 = eb * EB;

        v8f C[8] = {};   // accumulates T[:, e-tile] over the whole k dimension

        for (int ks = 0; ks < D / KS; ++ks) {
            __syncthreads();
            // Stage W[ks*KS .. +KS) x [e0 .. e0+EB) into LDS, transposed to [e][k].
            // Global side: 128 consecutive threads read 128 consecutive e (coalesced).
#pragma unroll 4
            for (int i = tid; i < KS * EB; i += 256) {
                int e  = i & (EB - 1);
                int kl = i >> 7;
                sW[e * KPAD + kl] = W[(size_t)(ks * KS + kl) * D + e0 + e];
            }
            __syncthreads();

#pragma unroll
            for (int kt = 0; kt < KS / 16; ++kt) {
#pragma unroll
                for (int j = 0; j < 4; ++j) {
                    const int klA = kt * 16 + j * 4 + kh;     // local k of this frag pair
                    // A frag (16x4 f32): lane&15 = M, {klA, klA+1} contiguous -> global b64
                    v2f a = *reinterpret_cast<const v2f*>(aRow + ks * KS + klA);
#pragma unroll
                    for (int t = 0; t < 8; ++t) {
                        // B frag: two consecutive floats in [e][k] layout -> one ds_load_b64
                        v2f b = *reinterpret_cast<const v2f*>(
                            &sW[(t * 16 + colg) * KPAD + klA]);
                        C[t] = __builtin_amdgcn_wmma_f32_16x16x4_f32(
                            false, a, false, b, (short)0, C[t], false, false);
                    }
                }
            }
        }

        // Fused epilogue for this e-block: acc[r] += C[t][r] * eth[M, col]
        // C layout: VGPR r = row (r | (lane>=16)*8), lane&15 = column.
        const int mOff = (lane >> 4) * 8;
#pragma unroll
        for (int t = 0; t < 8; ++t) {
            const int col = e0 + t * 16 + colg;
#pragma unroll
            for (int r = 0; r < 8; ++r) {
                float ev = eth[(size_t)(rowTile + mOff + r) * D + col];
                acc[r] = fmaf(C[t][r], ev, acc[r]);
            }
        }
    }

    // Reduce across the 16 column-lanes of each half-wave.
#pragma unroll
    for (int m = 1; m <= 8; m <<= 1) {
#pragma unroll
        for (int r = 0; r < 8; ++r) acc[r] += __shfl_xor(acc[r], m, 32);
    }

    if (colg == 0) {
        const int mOff = (lane >> 4) * 8;
#pragma unroll
        for (int r = 0; r < 8; ++r) out[rowTile + mOff + r] = acc[r];
    }
}

extern "C" void kernel_launch(void* const* d_in, const int* in_sizes, int n_in,
                              void* d_out, int out_size, void* d_ws, size_t ws_size,
                              hipStream_t stream) {
    const float* elg = (const float*)d_in[0];
    const float* eth = (const float*)d_in[1];
    const float* W   = (const float*)d_in[2];
    float* out = (float*)d_out;

    const int nRows = in_sizes[0] / D;                 // 32768
    dim3 grid(nRows / ROWS_PER_BLOCK);                 // 256 workgroups
    dim3 block(256);                                   // 8 wave32
    bilinear_wmma_f32_kernel<<<grid, block, 0, stream>>>(elg, eth, W, out);
}